// GCN_27023934226652
// MI455X (gfx1250) — compile-verified
//
#include <hip/hip_runtime.h>

typedef float v2f __attribute__((ext_vector_type(2)));
typedef float v8f __attribute__((ext_vector_type(8)));

#define NN 100000
#define NE 1600000

// ---------------------------------------------------------------- utilities
__global__ void zero_f32_kernel(float* __restrict__ p, long long n) {
  long long i = (long long)blockIdx.x * blockDim.x + threadIdx.x;
  if (i < n) p[i] = 0.0f;
}

__global__ void init_deg_kernel(float* __restrict__ deg, int n) {
  int i = blockIdx.x * blockDim.x + threadIdx.x;
  if (i < n) deg[i] = 1.0f;  // self-loop contributes 1 to every destination degree
}

__global__ void count_deg_kernel(const int* __restrict__ col, float* __restrict__ deg, int e) {
  int i = blockIdx.x * blockDim.x + threadIdx.x;
  if (i < e) atomicAdd(&deg[col[i]], 1.0f);
}

__global__ void make_dinv_kernel(const float* __restrict__ deg, float* __restrict__ dinv, int n) {
  int i = blockIdx.x * blockDim.x + threadIdx.x;
  if (i < n) dinv[i] = rsqrtf(deg[i]);  // deg >= 1 always (self-loops)
}

// ---------------------------------------------------------------- WMMA GEMM
// H[N x FOUT] = X[N x FIN] @ W[FIN x FOUT], full f32 via V_WMMA_F32_16X16X4_F32.
// One wave computes one 16x16 output tile; blockDim.x = 32 * (FOUT/16) so the
// FOUT/16 column tiles of a 16-row strip live in one block. N % 16 == 0, so
// EXEC is all-1s everywhere (WMMA requirement), no predication needed.
//
// ISA layouts (CDNA5 ISA §7.12.2), m = lane&15, hi = lane>>4:
//   A (16x4, MxK):  vgpr0 = A[m][2*hi],  vgpr1 = A[m][2*hi+1]
//   B (4x16, KxN):  vgpr0 = B[2*hi][n],  vgpr1 = B[2*hi+1][n]   (n = m)
//   C/D (16x16):    vgpr r = D[r + 8*hi][m]
template <int FIN, int FOUT>
__global__ __launch_bounds__(32 * (FOUT / 16)) void gemm_wmma_kernel(
    const float* __restrict__ X, const float* __restrict__ W, float* __restrict__ H) {
  const int lane = threadIdx.x & 31;
  const int wave = threadIdx.x >> 5;  // column tile within block
  const int m = lane & 15;
  const int hi = lane >> 4;

  const int row0 = blockIdx.x * 16;
  const int col0 = wave * 16;

  const float* xrow = X + (size_t)(row0 + m) * FIN;

  v8f acc = {};
#pragma unroll
  for (int kk = 0; kk < FIN; kk += 4) {
    const int ka = kk + 2 * hi;
    v2f a;
    a.x = xrow[ka];
    a.y = xrow[ka + 1];
    v2f b;
    b.x = W[(size_t)ka * FOUT + col0 + m];
    b.y = W[(size_t)(ka + 1) * FOUT + col0 + m];
    acc = __builtin_amdgcn_wmma_f32_16x16x4_f32(
        /*neg_a=*/false, a, /*neg_b=*/false, b,
        /*c_mod=*/(short)0, acc, /*reuse_a=*/false, /*reuse_b=*/false);
  }

#pragma unroll
  for (int r = 0; r < 8; ++r) {
    H[(size_t)(row0 + r + 8 * hi) * FOUT + col0 + m] = acc[r];
  }
}

// ------------------------------------------------------- edge scatter (L2 atomics)
// agg[col] += H[row] * dinv[row]*dinv[col].  FOUT/4 lanes per edge, each lane
// moves one float4 (128b load) and issues 4 global_atomic_add_f32 that resolve
// in the 192MB L2 (H + agg + indices ~= 64MB working set, fully L2-resident).
template <int F>
__global__ void scatter_edges_kernel(const float* __restrict__ H,
                                     const int* __restrict__ row_idx,
                                     const int* __restrict__ col_idx,
                                     const float* __restrict__ dinv,
                                     float* __restrict__ agg, int n_edges) {
  constexpr int LPE = F / 4;  // power of two (16 or 8)
  long long t = (long long)blockIdx.x * blockDim.x + threadIdx.x;
  int e = (int)(t / LPE);
  int sub = (int)(t % LPE);
  if (e >= n_edges) return;

  const int r = row_idx[e];
  const int c = col_idx[e];
  const float nrm = dinv[r] * dinv[c];

  const float4 hv = *(const float4*)(H + (size_t)r * F + sub * 4);
  float* o = agg + (size_t)c * F + sub * 4;
  atomicAdd(o + 0, hv.x * nrm);
  atomicAdd(o + 1, hv.y * nrm);
  atomicAdd(o + 2, hv.z * nrm);
  atomicAdd(o + 3, hv.w * nrm);
}

// ------------------------------------------------- self-loop + bias + ReLU fuse
template <int F, bool RELU>
__global__ void finalize_kernel(const float* __restrict__ agg, const float* __restrict__ H,
                                const float* __restrict__ dinv, const float* __restrict__ bias,
                                float* __restrict__ out, int n) {
  constexpr int V = F / 4;
  long long t = (long long)blockIdx.x * blockDim.x + threadIdx.x;
  if (t >= (long long)n * V) return;
  int i = (int)(t / V);
  int f = (int)(t % V) * 4;

  const float di = dinv[i];
  const float s = di * di;  // self-loop norm
  const float4 av = *(const float4*)(agg + (size_t)i * F + f);
  const float4 hv = *(const float4*)(H + (size_t)i * F + f);
  const float4 bv = *(const float4*)(bias + f);

  float4 v;
  v.x = av.x + hv.x * s + bv.x;
  v.y = av.y + hv.y * s + bv.y;
  v.z = av.z + hv.z * s + bv.z;
  v.w = av.w + hv.w * s + bv.w;
  if (RELU) {
    v.x = fmaxf(v.x, 0.0f);
    v.y = fmaxf(v.y, 0.0f);
    v.z = fmaxf(v.z, 0.0f);
    v.w = fmaxf(v.w, 0.0f);
  }
  *(float4*)(out + (size_t)i * F + f) = v;
}

// ---------------------------------------------------------------- one GCN layer
template <int FIN, int FOUT, bool RELU>
static void run_layer(const float* X, const float* W, const float* b, float* out,
                      float* h, float* agg, const float* dinv,
                      const int* row, const int* col, hipStream_t stream) {
  gemm_wmma_kernel<FIN, FOUT><<<NN / 16, 32 * (FOUT / 16), 0, stream>>>(X, W, h);

  const long long agg_n = (long long)NN * FOUT;
  zero_f32_kernel<<<(int)((agg_n + 255) / 256), 256, 0, stream>>>(agg, agg_n);

  const long long sc_threads = (long long)NE * (FOUT / 4);
  scatter_edges_kernel<FOUT><<<(int)((sc_threads + 255) / 256), 256, 0, stream>>>(
      h, row, col, dinv, agg, NE);

  const long long fin_threads = (long long)NN * (FOUT / 4);
  finalize_kernel<FOUT, RELU><<<(int)((fin_threads + 255) / 256), 256, 0, stream>>>(
      agg, h, dinv, b, out, NN);
}

// ---------------------------------------------------------------- entry point
extern "C" void kernel_launch(void* const* d_in, const int* in_sizes, int n_in,
                              void* d_out, int out_size, void* d_ws, size_t ws_size,
                              hipStream_t stream) {
  (void)in_sizes; (void)n_in; (void)out_size; (void)ws_size;

  const float* x  = (const float*)d_in[0];
  const int*   ei = (const int*)d_in[1];
  const float* W1 = (const float*)d_in[2];
  const float* b1 = (const float*)d_in[3];
  const float* W2 = (const float*)d_in[4];
  const float* b2 = (const float*)d_in[5];
  const float* W3 = (const float*)d_in[6];
  const float* b3 = (const float*)d_in[7];
  const float* W4 = (const float*)d_in[8];
  const float* b4 = (const float*)d_in[9];

  const int* row = ei;        // edge_index[0] = source
  const int* col = ei + NE;   // edge_index[1] = destination

  // workspace carve-up (256B aligned); everything below is fully re-written
  // every call, so 0xAA poison is harmless.
  char* ws = (char*)d_ws;
  size_t off = 0;
  auto carve = [&](size_t bytes) -> float* {
    float* p = (float*)(ws + off);
    off = (off + bytes + 255) & ~(size_t)255;
    return p;
  };
  float* deg  = carve((size_t)NN * 4);
  float* dinv = carve((size_t)NN * 4);
  float* h    = carve((size_t)NN * 64 * 4);
  float* agg  = carve((size_t)NN * 64 * 4);
  float* bufA = carve((size_t)NN * 64 * 4);
  float* bufB = carve((size_t)NN * 64 * 4);

  // symmetric normalization coefficients
  init_deg_kernel<<<(NN + 255) / 256, 256, 0, stream>>>(deg, NN);
  count_deg_kernel<<<(NE + 255) / 256, 256, 0, stream>>>(col, deg, NE);
  make_dinv_kernel<<<(NN + 255) / 256, 256, 0, stream>>>(deg, dinv, NN);

  // 4 GCN layers (ReLU on 1-3, none on 4)
  run_layer<128, 64, true >(x,    W1, b1, bufA,          h, agg, dinv, row, col, stream);
  run_layer< 64, 64, true >(bufA, W2, b2, bufB,          h, agg, dinv, row, col, stream);
  run_layer< 64, 64, true >(bufB, W3, b3, bufA,          h, agg, dinv, row, col, stream);
  run_layer< 64, 32, false>(bufA, W4, b4, (float*)d_out, h, agg, dinv, row, col, stream);
}